// Conv2D_Dense_51926154609342
// MI455X (gfx1250) — compile-verified
//
#include <hip/hip_runtime.h>
#include <hip/hip_bf16.h>

typedef __attribute__((ext_vector_type(2))) float v2f;
typedef __attribute__((ext_vector_type(8))) float v8f;

#define IW 4096
#define IH 4096
#define OW 4088
#define TILES_X 256   // ceil(4088/16)
#define TILES_Y 256
#define STRIPS_Y 64   // 4 tiles (64 rows) per wave job
#define WAVES_PER_BLOCK 8

// ---------------- Edge path: one 16x16 tile, clamped loads, guarded stores ----
__device__ __forceinline__ void conv_tile_edge(
    const float* __restrict__ X,
    const float (&skern)[81],
    float* __restrict__ out,
    float b0, int x0, int y0, int lane)
{
    const int n  = lane & 15;
    const int kb = (lane >> 4) << 1;

    v8f acc = {};

    #pragma unroll
    for (int c = 0; c < 6; ++c) {
        const int dxb = 4 * c + kb;
        const int dd  = dxb - n;
        const bool m0 = (dd     >= 0) && (dd     <= 8);
        const bool m1 = (dd + 1 >= 0) && (dd + 1 <= 8);
        const int  i0 = min(max(dd,     0), 8);
        const int  i1 = min(max(dd + 1, 0), 8);

        #pragma unroll
        for (int ky = 0; ky < 9; ++ky) {
            int row = y0 + n + ky;
            row = row < IH - 1 ? row : IH - 1;            // only hits invalid out rows
            int idx = row * IW + (x0 + dxb);
            idx = idx < IW * IW - 2 ? idx : IW * IW - 2;  // only hits invalid out cols
            v2f a = *(const v2f*)(X + idx);

            float t0 = skern[ky * 9 + i0];
            float t1 = skern[ky * 9 + i1];
            v2f b;
            b.x = m0 ? t0 : 0.0f;
            b.y = m1 ? t1 : 0.0f;

            acc = __builtin_amdgcn_wmma_f32_16x16x4_f32(
                false, a, false, b, (short)0, acc, false, false);
        }
    }

    const int mbase = (lane >> 4) << 3;
    const int ox = x0 + n;
    #pragma unroll
    for (int i = 0; i < 8; ++i) {
        const int oy = y0 + mbase + i;
        if (oy < OW && ox < OW) out[oy * OW + ox] = acc[i] + b0;
    }
}

// ---------------- Fast path: 16x64 strip (4 tiles), shared B fragments --------
__device__ __forceinline__ void conv_strip4(
    const float* __restrict__ X,
    const float (&skern)[81],
    float* __restrict__ out,
    float b0, int x0, int y0, int lane)
{
    const int n  = lane & 15;
    const int kb = (lane >> 4) << 1;

    v8f acc[4] = {{}, {}, {}, {}};

    // All 216 A-loads: one base, compile-time immediate offsets (< 1 MB).
    const float* base = X + (y0 + n) * IW + (x0 + kb);

    #pragma unroll
    for (int c = 0; c < 6; ++c) {
        const int dxb = 4 * c + kb;
        const int dd  = dxb - n;
        const bool m0 = (dd     >= 0) && (dd     <= 8);
        const bool m1 = (dd + 1 >= 0) && (dd + 1 <= 8);
        const int  i0 = min(max(dd,     0), 8);
        const int  i1 = min(max(dd + 1, 0), 8);

        #pragma unroll
        for (int ky = 0; ky < 9; ++ky) {
            // B fragment built ONCE, reused by 4 sub-tiles (amortized ds + cndmask)
            float t0 = skern[ky * 9 + i0];
            float t1 = skern[ky * 9 + i1];
            v2f b;
            b.x = m0 ? t0 : 0.0f;
            b.y = m1 ? t1 : 0.0f;

            #pragma unroll
            for (int s = 0; s < 4; ++s) {
                v2f a = *(const v2f*)(base + (s * 16 + ky) * IW + 4 * c);
                acc[s] = __builtin_amdgcn_wmma_f32_16x16x4_f32(
                    false, a, false, b, (short)0, acc[s], false, false);
            }
        }
    }

    // Stores: one base, immediate offsets (< 1 MB).
    const int mbase = (lane >> 4) << 3;
    float* orow = out + (y0 + mbase) * OW + (x0 + n);
    #pragma unroll
    for (int s = 0; s < 4; ++s) {
        #pragma unroll
        for (int i = 0; i < 8; ++i) {
            orow[(s * 16 + i) * OW] = acc[s][i] + b0;
        }
    }
}

__global__ __launch_bounds__(256) void conv9x9_wmma_f32(
    const float* __restrict__ X,
    const float* __restrict__ Kf,
    const float* __restrict__ bias,
    float* __restrict__ out)
{
    __shared__ float skern[81];
    for (int t = threadIdx.x; t < 81; t += 256) skern[t] = Kf[t];
    __syncthreads();

    const int lane = threadIdx.x & 31;
    const int wave = threadIdx.x >> 5;
    const int job  = blockIdx.x * WAVES_PER_BLOCK + wave;   // 64 x 256 strip jobs
    const int tx   = job & (TILES_X - 1);
    const int sy   = job >> 8;                              // 0..63
    const int x0   = tx * 16;
    const int y0   = sy * 64;

    const float b0 = bias[0];

    // Wave-uniform branch: EXEC stays all-1s in both arms (WMMA requirement).
    if (tx < TILES_X - 1 && sy < STRIPS_Y - 1) {
        conv_strip4(X, skern, out, b0, x0, y0, lane);       // 98.1% of jobs
    } else {
        #pragma unroll
        for (int s = 0; s < 4; ++s) {
            conv_tile_edge(X, skern, out, b0, x0, y0 + s * 16, lane);
        }
    }
}

extern "C" void kernel_launch(void* const* d_in, const int* in_sizes, int n_in,
                              void* d_out, int out_size, void* d_ws, size_t ws_size,
                              hipStream_t stream) {
    const float* X    = (const float*)d_in[0];
    const float* Kf   = (const float*)d_in[1];
    const float* bias = (const float*)d_in[2];
    float* out        = (float*)d_out;

    const int total_jobs = TILES_X * STRIPS_Y;              // 16384 strip jobs
    const int blocks = total_jobs / WAVES_PER_BLOCK;        // 2048

    hipLaunchKernelGGL(conv9x9_wmma_f32, dim3(blocks), dim3(256), 0, stream,
                       X, Kf, bias, out);
}